// DP_CAML_33646773797448
// MI455X (gfx1250) — compile-verified
//
#include <hip/hip_runtime.h>

typedef __attribute__((ext_vector_type(2))) float v2f;
typedef __attribute__((ext_vector_type(8))) float v8f;

#define VOCAB 50000
#define LROWS 8921
#define DCH   300
#define DP2   150       // DCH/2 pairs
#define KF    10
#define BB    8
#define TLEN  2048
#define TP    2057      // TLEN + KF - 1
#define PADK  9         // KF - 1

#define XCNT  (BB * DCH * TLEN)   // floats (stored as XCNT/2 float2 pairs)
#define HCNT  (BB * DCH * TP)     // floats (stored pair-interleaved)

// ------------- Kernel 0: embedding gather -> x2[b][d/2][t] (pair-interleaved)
__global__ __launch_bounds__(256)
void DP_gather(const int* __restrict__ ids, const float* __restrict__ embed,
               v2f* __restrict__ x2) {
  int idx = blockIdx.x * blockDim.x + threadIdx.x;   // over (b, d/2, t)
  if (idx >= XCNT / 2) return;
  int t = idx % TLEN;
  int p = (idx / TLEN) % DP2;
  int b = idx / (TLEN * DP2);
  int row = ids[b * TLEN + t];
  x2[idx] = *(const v2f*)&embed[row * DCH + 2 * p];  // 8B aligned (300*4*row + 8p)
}

// ------------- Kernel 1: conv1d (pad=9) + bias + relu via WMMA --------------
// h[b,o,t'] = relu(bias[o] + sum_{i,k} x[b,i,t'-9+k] * w[o,i,k])
// A = w (16 o x 4 i-slots), B = x (4 i-slots x 16 t').  All frags = 1x b64.
#define TCHUNK 512
#define XJ     521          // TCHUNK + KF - 1 columns
#define XSTR2  528          // float2 row stride: 2*528 % 64 == 32 -> halves split banks
#define WCNT2  (KF * 8 * 16)  // k * ii2 * o  float2 elements

__global__ __launch_bounds__(256, 1)
void DP_conv(const v2f* __restrict__ x2, const float* __restrict__ conv_w,
             const float* __restrict__ conv_b, float* __restrict__ h2f) {
  __shared__ v2f wl2t[WCNT2];        // [(k*8+ii2)*16 + o]  (transposed: lane==o)
  __shared__ v2f xl2[8 * XSTR2];     // [ii2][col]

  const int b   = blockIdx.z;
  const int o0  = blockIdx.y * 16;
  const int t0  = blockIdx.x * TCHUNK;
  const int tid = threadIdx.x;
  const int lane = tid & 31;
  const int wid  = tid >> 5;
  const int m    = lane & 15;
  const int lh   = lane >> 4;

  v8f c0 = {0.f,0.f,0.f,0.f,0.f,0.f,0.f,0.f};
  v8f c1 = c0, c2 = c0, c3 = c0;

  for (int i0 = 0; i0 < DCH; i0 += 16) {
    __syncthreads();
    // stage w chunk (transposed): wl2t[(k*8+ii2)*16+o] = w[o0+o][i0+2*ii2 .. +1][k]
    for (int idx = tid; idx < WCNT2; idx += 256) {
      int o = idx & 15, s = idx >> 4;      // s = k*8 + ii2
      int k = s >> 3, ii2 = s & 7;
      int oo = o0 + o, i = i0 + 2 * ii2;
      v2f v = {0.f, 0.f};
      if (oo < DCH && i < DCH) {
        const float* wp = &conv_w[(oo * DCH + i) * KF + k];
        v.x = wp[0]; v.y = wp[KF];         // i and i+1 (i even, i+1 <= 299)
      }
      wl2t[idx] = v;
    }
    // stage x chunk: xl2[ii2][j] = x2[b][i0/2+ii2][t0-9+j], zero outside [0,TLEN)
    for (int idx = tid; idx < 8 * XJ; idx += 256) {
      int ii2 = idx / XJ, j = idx % XJ;
      int i2 = i0 / 2 + ii2, tau = t0 - PADK + j;
      v2f v = {0.f, 0.f};
      if (i2 < DP2 && tau >= 0 && tau < TLEN) v = x2[(b * DP2 + i2) * TLEN + tau];
      xl2[ii2 * XSTR2 + j] = v;
    }
    __syncthreads();

    for (int pb = 0; pb < 8; pb += 2) {            // pair-base over i-chunk
      for (int k = 0; k < KF; ++k) {
        v2f a = wl2t[(k * 8 + pb + lh) * 16 + m];  // 1 ds_b64, conflict-free
        const v2f* xr = &xl2[(pb + lh) * XSTR2 + wid * 64 + m + k];
        c0 = __builtin_amdgcn_wmma_f32_16x16x4_f32(false, a, false, xr[0],  (short)0, c0, false, false);
        c1 = __builtin_amdgcn_wmma_f32_16x16x4_f32(false, a, false, xr[16], (short)0, c1, false, false);
        c2 = __builtin_amdgcn_wmma_f32_16x16x4_f32(false, a, false, xr[32], (short)0, c2, false, false);
        c3 = __builtin_amdgcn_wmma_f32_16x16x4_f32(false, a, false, xr[48], (short)0, c3, false, false);
      }
    }
  }

  // epilogue: bias + relu; store pair-interleaved h2[b][o/2][t'][o&1]
  auto emit = [&](const v8f& c, int n) {
    int t = t0 + wid * 64 + n * 16 + m;
    if (t >= TP) return;
#pragma unroll
    for (int e = 0; e < 8; ++e) {
      int orow = o0 + e + 8 * lh;                  // C layout: row = e + 8*(lane>=16)
      if (orow < DCH) {
        float v = c[e] + conv_b[orow];
        h2f[((b * DP2 + (orow >> 1)) * TP + t) * 2 + (orow & 1)] = v > 0.f ? v : 0.f;
      }
    }
  };
  emit(c0, 0); emit(c1, 1); emit(c2, 2); emit(c3, 3);
}

// ------------- Kernel 2: scores GEMM + fused online softmax -> logits -------
// s[b,l,t] = sum_d U[l,d]*h[b,d,t];  logits[b,l] = sum_t softmax_t(s)*s + bias
#define LT   558            // ceil(8921/16)
#define NTT  129            // ceil(2057/16)

__global__ __launch_bounds__(256, 1)
void DP_attn(const v2f* __restrict__ h2, const float* __restrict__ U,
             const float* __restrict__ fc_bias, float* __restrict__ out) {
  __shared__ v2f Ul2t[DP2 * 16];     // [d2*16 + r] (transposed: lane==r), no pad
  __shared__ float redm[2048], redz[2048], reds[2048];

  const int b   = blockIdx.y;
  const int l0  = blockIdx.x * 16;
  const int tid = threadIdx.x;
  const int lane = tid & 31, wid = tid >> 5;
  const int m = lane & 15, lh = lane >> 4;

  for (int idx = tid; idx < DP2 * 16; idx += 256) {
    int r = idx & 15, d2 = idx >> 4;
    v2f v = {0.f, 0.f};
    if (l0 + r < LROWS) v = *(const v2f*)&U[(l0 + r) * DCH + 2 * d2];
    Ul2t[idx] = v;
  }
  __syncthreads();

  float am[8], az[8], as_[8];
#pragma unroll
  for (int e = 0; e < 8; ++e) { am[e] = -1e30f; az[e] = 0.f; as_[e] = 0.f; }

  const v2f* hb2 = h2 + (size_t)b * DP2 * TP;
  for (int tt = wid * 2; tt < NTT; tt += 16) {     // 2 t-tiles per A-fragment
    const int ta = tt * 16 + m;
    const int tbt = ta + 16;
    const bool va = ta < TP;
    const bool vb = tbt < TP;
    const int tca = va ? ta : (TP - 1);
    const int tcb = vb ? tbt : (TP - 1);
    v8f c0 = {0.f,0.f,0.f,0.f,0.f,0.f,0.f,0.f};
    v8f c1 = c0;
    const v2f* hpa = hb2 + (size_t)lh * TP + tca;
    const v2f* hpb = hb2 + (size_t)lh * TP + tcb;
    const v2f* up  = &Ul2t[lh * 16 + m];
    for (int d0 = 0; d0 < DCH; d0 += 4) {
      v2f a = up[0]; up += 32;                     // pair d0/2+lh -> next step +2 pairs
      c0 = __builtin_amdgcn_wmma_f32_16x16x4_f32(false, a, false, hpa[0], (short)0, c0, false, false);
      c1 = __builtin_amdgcn_wmma_f32_16x16x4_f32(false, a, false, hpb[0], (short)0, c1, false, false);
      hpa += 2 * TP; hpb += 2 * TP;
    }
    auto absorb = [&](const v8f& c, bool valid) {
      if (!valid) return;
#pragma unroll
      for (int e = 0; e < 8; ++e) {
        float v  = c[e];
        float nm = fmaxf(am[e], v);
        float p  = __expf(am[e] - nm);
        float q  = __expf(v - nm);
        az[e]  = az[e]  * p + q;
        as_[e] = as_[e] * p + v * q;
        am[e]  = nm;
      }
    };
    absorb(c0, va);
    absorb(c1, vb);
  }

  // merge: 16 lanes per row per wave, then 8 waves, via LDS
#pragma unroll
  for (int e = 0; e < 8; ++e) {
    int row = e + 8 * lh;
    int idx = (wid * 16 + row) * 16 + m;
    redm[idx] = am[e]; redz[idx] = az[e]; reds[idx] = as_[e];
  }
  __syncthreads();
  if (tid < 16) {
    int row = tid;
    float M = -1e30f, Z = 0.f, S = 0.f;
    for (int w = 0; w < 8; ++w)
      for (int s = 0; s < 16; ++s) {
        int idx = (w * 16 + row) * 16 + s;
        float mi = redm[idx];
        float nm = fmaxf(M, mi);
        float p = __expf(M - nm), q = __expf(mi - nm);
        Z = Z * p + redz[idx] * q;
        S = S * p + reds[idx] * q;
        M = nm;
      }
    int l = l0 + row;
    if (l < LROWS) out[b * LROWS + l] = S / Z + fc_bias[l];
  }
}

extern "C" void kernel_launch(void* const* d_in, const int* in_sizes, int n_in,
                              void* d_out, int out_size, void* d_ws, size_t ws_size,
                              hipStream_t stream) {
  const int*   ids     = (const int*)d_in[0];
  const float* embed_w = (const float*)d_in[1];
  const float* conv_w  = (const float*)d_in[2];
  const float* conv_b  = (const float*)d_in[3];
  const float* U       = (const float*)d_in[4];
  const float* fc_bias = (const float*)d_in[5];
  float* out = (float*)d_out;

  v2f* x2 = (v2f*)d_ws;                     // XCNT/2 pairs
  v2f* h2 = x2 + XCNT / 2;                  // HCNT/2 pairs; ws needs ~39.4 MB
  float* h2f = (float*)h2;

  DP_gather<<<(XCNT / 2 + 255) / 256, 256, 0, stream>>>(ids, embed_w, x2);
  DP_conv<<<dim3((TP + TCHUNK - 1) / TCHUNK, (DCH + 15) / 16, BB), 256, 0, stream>>>(
      x2, conv_w, conv_b, h2f);
  DP_attn<<<dim3(LT, BB), 256, 0, stream>>>(h2, U, fc_bias, out);
}